// DropGlobalScaledDotProductAttention_23854248362057
// MI455X (gfx1250) — compile-verified
//
#include <hip/hip_runtime.h>

typedef float v2f __attribute__((ext_vector_type(2)));
typedef float v8f __attribute__((ext_vector_type(8)));

#define LQ   512
#define DK   64
#define DDIM 256
#define F2   512        // 2*DD
#define NEGV (-1e9f)
#define INV_TEMP 0.125f // 1/sqrt(dk)

// ---------------------------------------------------------------------------
// Phase 1: a[b,i,f] = d1[b,i,:]·W1[:256,f]   (g&1==0)
//          c[b,j,f] = d0[b,j,:]·W1[256:,f]   (g&1==1)
// One wave per 16x16 tile, fp32 WMMA 16x16x4, K=256 (64 chained WMMAs).
// f32 A layout: lane(h=lane>>4, m=lane&15): VGPR0=K(2h), VGPR1=K(2h+1), row M=m
// f32 B layout (C/D-pattern split): VGPR0=K(2h), VGPR1=K(2h+1), col N=lane&15
// ---------------------------------------------------------------------------
__global__ __launch_bounds__(32)
void ac_gemm_kernel(const float* __restrict__ d0, const float* __restrict__ d1,
                    const float* __restrict__ W1,
                    float* __restrict__ a_out, float* __restrict__ c_out) {
  const int m0 = blockIdx.x * 16;
  const int n0 = blockIdx.y * 16;
  const int g  = blockIdx.z;          // 0..3 : {batch, half}
  const int bb = g >> 1;
  const int hh = g & 1;

  const float* src = (hh == 0 ? d1 : d0) + (size_t)bb * LQ * DDIM;
  const float* W   = W1 + (size_t)hh * DDIM * F2;   // key-half uses rows 256..511
  float*       dst = (hh == 0 ? a_out : c_out) + (size_t)bb * LQ * F2;

  const int lane = threadIdx.x;
  const int lh   = lane >> 4;         // 0/1 half of wave
  const int m    = lane & 15;

  const float* arow = src + (size_t)(m0 + m) * DDIM;
  const int    ncol = n0 + m;

  v8f acc = {};
  for (int k0 = 0; k0 < DDIM; k0 += 4) {
    const int kk = k0 + 2 * lh;
    v2f av = *(const v2f*)(arow + kk);
    v2f bv;
    bv.x = W[(size_t)(kk + 0) * F2 + ncol];
    bv.y = W[(size_t)(kk + 1) * F2 + ncol];
    acc = __builtin_amdgcn_wmma_f32_16x16x4_f32(false, av, false, bv,
                                                (short)0, acc, false, false);
  }
  // C/D layout: VGPR v -> row m0 + 8*lh + v, col ncol
  float* orow = dst + (size_t)(m0 + 8 * lh) * F2 + ncol;
#pragma unroll
  for (int v = 0; v < 8; ++v) orow[(size_t)v * F2] = acc[v];
}

// ---------------------------------------------------------------------------
// Phase 2: decisions[b,i,j] = (logit1 > logit0) ? -1e9 : 0
//   logit_o = sum_f relu(a[i,f] + c[j,f] + b1[f]) * W2[f,o] + b2[o]
// Straight-through y == y_hard exactly in fp32 (Sterbenz), so hard threshold.
// 16x16 pair tile per 256-thread block; f processed in two 256-chunks in LDS.
// ---------------------------------------------------------------------------
#define FCH 256
__global__ __launch_bounds__(256)
void decide_kernel(const float* __restrict__ a_in, const float* __restrict__ c_in,
                   const float* __restrict__ b1, const float* __restrict__ W2,
                   const float* __restrict__ b2, float* __restrict__ dec) {
  __shared__ float a_s[16][FCH + 1];   // stride 257: bank = (r*257+f)%64 conflict-free
  __shared__ float c_s[16][FCH + 1];
  __shared__ float w2_s[2 * F2];       // interleaved (w0,w1) pairs

  const int bb = blockIdx.z;
  const int i0 = blockIdx.y * 16;
  const int j0 = blockIdx.x * 16;
  const float* A = a_in + ((size_t)bb * LQ + i0) * F2;
  const float* C = c_in + ((size_t)bb * LQ + j0) * F2;

  const int tid = threadIdx.x;
  for (int f = tid; f < F2; f += 256) {
    w2_s[2 * f + 0] = W2[2 * f + 0];
    w2_s[2 * f + 1] = W2[2 * f + 1];
  }

  const int tj = tid & 15;
  const int ti = tid >> 4;
  float l0 = b2[0];
  float l1 = b2[1];

  for (int ch = 0; ch < 2; ++ch) {
    const int fb = ch * FCH;
    __syncthreads();   // protect previous chunk's reads
    for (int idx = tid; idx < 16 * FCH; idx += 256) {
      const int r = idx >> 8;
      const int f = idx & (FCH - 1);
      a_s[r][f] = A[(size_t)r * F2 + fb + f];
      c_s[r][f] = C[(size_t)r * F2 + fb + f] + b1[fb + f];  // fold bias once
    }
    __syncthreads();
#pragma unroll 8
    for (int f = 0; f < FCH; ++f) {
      float t = a_s[ti][f] + c_s[tj][f];
      t = fmaxf(t, 0.0f);
      const v2f w = *(const v2f*)&w2_s[2 * (fb + f)];
      l0 = fmaf(t, w.x, l0);
      l1 = fmaf(t, w.y, l1);
    }
  }
  dec[((size_t)bb * LQ + i0 + ti) * LQ + j0 + tj] = (l1 > l0) ? NEGV : 0.0f;
}

// ---------------------------------------------------------------------------
// Phase 3: attn[b,n,i,j] = (q[b,n,i,:]·k[b,n,j,:]) * 0.125 + dec[b,i,j]
// NT GEMM: A = q rows (K contiguous), B = k^T so lane's B frag is a contiguous
// float2 from k row j. One wave per 16x16 tile, 16 chained f32 WMMAs (K=64).
// ---------------------------------------------------------------------------
__global__ __launch_bounds__(32)
void attn_kernel(const float* __restrict__ q, const float* __restrict__ kmat,
                 const float* __restrict__ dec, float* __restrict__ out) {
  const int j0 = blockIdx.x * 16;
  const int i0 = blockIdx.y * 16;
  const int z  = blockIdx.z;          // b*8 + n
  const int bb = z >> 3;

  const float* Q = q    + (size_t)z * LQ * DK;
  const float* K = kmat + (size_t)z * LQ * DK;

  const int lane = threadIdx.x;
  const int lh   = lane >> 4;
  const int m    = lane & 15;

  const float* qrow = Q + (size_t)(i0 + m) * DK;
  const float* krow = K + (size_t)(j0 + m) * DK;

  v8f acc = {};
#pragma unroll
  for (int k0 = 0; k0 < DK; k0 += 4) {
    const int kk = k0 + 2 * lh;
    v2f av = *(const v2f*)(qrow + kk);
    v2f bv = *(const v2f*)(krow + kk);
    acc = __builtin_amdgcn_wmma_f32_16x16x4_f32(false, av, false, bv,
                                                (short)0, acc, false, false);
  }

  const float* drow = dec + ((size_t)bb * LQ + i0 + 8 * lh) * LQ + j0 + m;
  float*       orow = out + ((size_t)z  * LQ + i0 + 8 * lh) * LQ + j0 + m;
#pragma unroll
  for (int v = 0; v < 8; ++v)
    orow[(size_t)v * LQ] = acc[v] * INV_TEMP + drow[(size_t)v * LQ];
}

// ---------------------------------------------------------------------------
extern "C" void kernel_launch(void* const* d_in, const int* in_sizes, int n_in,
                              void* d_out, int out_size, void* d_ws, size_t ws_size,
                              hipStream_t stream) {
  (void)in_sizes; (void)n_in; (void)out_size; (void)ws_size;
  const float* q  = (const float*)d_in[0];
  const float* k  = (const float*)d_in[1];
  const float* d0 = (const float*)d_in[2];
  const float* d1 = (const float*)d_in[3];
  const float* W1 = (const float*)d_in[4];
  const float* b1 = (const float*)d_in[5];
  const float* W2 = (const float*)d_in[6];
  const float* b2 = (const float*)d_in[7];
  float* out = (float*)d_out;

  float* a_ws = (float*)d_ws;                        // [2,512,512] = 2 MB
  float* c_ws = a_ws + (size_t)2 * LQ * F2;          // [2,512,512] = 2 MB
  float* dcs  = c_ws + (size_t)2 * LQ * F2;          // [2,512,512] = 2 MB

  ac_gemm_kernel<<<dim3(32, 32, 4),  32,  0, stream>>>(d0, d1, W1, a_ws, c_ws);
  decide_kernel <<<dim3(32, 32, 2),  256, 0, stream>>>(a_ws, c_ws, b1, W2, b2, dcs);
  attn_kernel   <<<dim3(32, 32, 16), 32,  0, stream>>>(q, k, dcs, out);
}